// SpliceFormerBlock_26345329394034
// MI455X (gfx1250) — compile-verified
//
#include <hip/hip_runtime.h>
#include <hip/hip_bf16.h>

#define SEQ   45000
#define BATCH 2
#define NCH   32
#define NSEL  512
#define HALF  256
#define HEADS 4
#define DK    32
#define HD    128
#define FEAT  30
#define MLPD  512
#define TBL   (2*SEQ - 1)

typedef __attribute__((ext_vector_type(2))) float v2f;
typedef __attribute__((ext_vector_type(8))) float v8f;
typedef __attribute__((ext_vector_type(4))) unsigned int u32x4;
typedef __attribute__((ext_vector_type(8))) int i32x8;
typedef __attribute__((ext_vector_type(4))) int i32x4;

__device__ __forceinline__ float warp_sum32(float v) {
#pragma unroll
  for (int m = 16; m >= 1; m >>= 1) v += __shfl_xor(v, m, 32);
  return v;
}

// ---------------------------------------------------------------- zero output
__global__ void zero_out_kernel(float* __restrict__ out, int n) {
  int t = blockIdx.x * blockDim.x + threadIdx.x;
  if (t < n) out[t] = 0.0f;
}

// ------------------------------------------------- top-256 per (batch, class)
// Rank = #elements with (value greater) or (equal value, smaller index):
// exactly matches stable argsort of -value. Ranks are a permutation, so
// exactly 256 elements get rank < 256 and each selected slot is written once.
__global__ void topk_rank_kernel(const float* __restrict__ attention,
                                 int* __restrict__ selraw) {
  __shared__ float sh[256];
  const int tid = threadIdx.x;
  const int b   = blockIdx.z;
  const int cls = blockIdx.y + 1;                 // classes 1 (acc), 2 (don)
  const float* av = attention + ((size_t)b * 3 + cls) * SEQ;
  const int pos = blockIdx.x * 256 + tid;
  const float my = (pos < SEQ) ? av[pos] : -3.4e38f;
  int cnt = 0;
  for (int base = 0; base < SEQ; base += 256) {
    int p = base + tid;
    sh[tid] = (p < SEQ) ? av[p] : -3.4e38f;
    __syncthreads();
    int lim = min(256, SEQ - base);
    for (int e = 0; e < lim; ++e) {
      float vv = sh[e];
      cnt += (vv > my) || (vv == my && (base + e) < pos);
    }
    __syncthreads();
  }
  if (pos < SEQ && cnt < HALF)
    selraw[b * NSEL + blockIdx.y * HALF + cnt] = pos;
}

// -------------------------------------------- bitonic sort of 512 ints (asc)
__global__ void sort_idx_kernel(const int* __restrict__ selraw,
                                int* __restrict__ idx) {
  __shared__ int s[NSEL];
  const int tid = threadIdx.x, b = blockIdx.x;
  s[tid] = selraw[b * NSEL + tid];
  __syncthreads();
  for (int k = 2; k <= NSEL; k <<= 1)
    for (int j = k >> 1; j > 0; j >>= 1) {
      int ixj = tid ^ j;
      if (ixj > tid) {
        int x = s[tid], y = s[ixj];
        bool up = ((tid & k) == 0);
        if ((x > y) == up) { s[tid] = y; s[ixj] = x; }
      }
      __syncthreads();
    }
  idx[b * NSEL + tid] = s[tid];
}

// -------------------------------------- Enformer positional basis table (f32)
__global__ void pos_table_kernel(float* __restrict__ tab) {
  int t = blockIdx.x * blockDim.x + threadIdx.x;
  if (t >= TBL) return;
  float d = (float)(t - (SEQ - 1));
  float a = fabsf(d);
  float sgn = (d > 0.f) ? 1.f : ((d < 0.f) ? -1.f : 0.f);
  float e[15];
  const float log2L = log2f((float)SEQ);
#pragma unroll
  for (int j = 0; j < 5; ++j) {           // exponential decay
    float ex = 3.0f + (float)j * (log2L - 3.0f) * 0.25f;
    float hl = exp2f(ex);
    e[j] = exp2f(-a / hl);
  }
#pragma unroll
  for (int j = 0; j < 5; ++j) {           // central mask
    float cw = (float)((2 << j) - 1);
    e[5 + j] = (cw > a) ? 1.0f : 0.0f;
  }
  float gmax = 0.0f;                      // gamma pdf, normalized by grid max
#pragma unroll
  for (int j = 0; j < 5; ++j) {
    float mean = 9000.0f + 9000.0f * (float)j;
    float cr   = mean / 4500.0f;
    float conc = cr * cr;
    float rate = mean / (4500.0f * 4500.0f);
    float logz = lgammaf(conc) - conc * logf(rate);
    float p = expf((conc - 1.0f) * logf(a) - rate * a - logz) + 1e-8f;
    e[10 + j] = p;
    float mode = (conc - 1.0f) / rate;    // pdf unimodal -> discrete max at
    float m0 = fminf(fmaxf(floorf(mode), 0.0f), (float)(SEQ - 1));
    float m1 = fminf(fmaxf(ceilf(mode),  0.0f), (float)(SEQ - 1));
    float p0 = expf((conc - 1.0f) * logf(m0) - rate * m0 - logz) + 1e-8f;
    float p1 = expf((conc - 1.0f) * logf(m1) - rate * m1 - logz) + 1e-8f;
    gmax = fmaxf(gmax, fmaxf(p0, p1));
  }
  float inv = 1.0f / gmax;
#pragma unroll
  for (int j = 0; j < 5; ++j) e[10 + j] *= inv;
  float* row = tab + (size_t)t * FEAT;
#pragma unroll
  for (int f = 0; f < 15; ++f) { row[f] = e[f]; row[15 + f] = sgn * e[f]; }
}

// ---------------------------------------------------------------- gather rows
__global__ void gather_emb_kernel(const float* __restrict__ x,
                                  const int* __restrict__ idx,
                                  float* __restrict__ emb) {
  int t = blockIdx.x * blockDim.x + threadIdx.x;
  if (t >= BATCH * NSEL * NCH) return;
  int c = t & 31, i = (t >> 5) & 511, b = t >> 14;
  int p = idx[b * NSEL + i];
  emb[t] = x[((size_t)b * NCH + c) * SEQ + p];
}

// ------------------------------------------------------ layernorm over C = 32
__global__ void layernorm_kernel(const float* __restrict__ x,
                                 const float* __restrict__ g,
                                 const float* __restrict__ bt,
                                 float* __restrict__ y, int rows) {
  int lane = threadIdx.x & 31;
  int r = blockIdx.x * (blockDim.x >> 5) + (threadIdx.x >> 5);
  if (r >= rows) return;
  float v   = x[r * NCH + lane];
  float mu  = warp_sum32(v) * (1.0f / 32.0f);
  float dlt = v - mu;
  float var = warp_sum32(dlt * dlt) * (1.0f / 32.0f);
  y[r * NCH + lane] = dlt * rsqrtf(var + 1e-5f) * g[lane] + bt[lane];
}

// ------------- generic f32 WMMA GEMM: C = act(A@B + bias) + Csrc (optional)
// One wave per 16x16 tile; K stepped by 4 via V_WMMA_F32_16X16X4_F32.
// A 16x4 layout: lanes 0-15 -> K=0,1 ; lanes 16-31 -> K=2,3 (per VGPR pair).
// B 4x16 layout mirrors with N = lane%16. C/D: VGPR v, M = v + 8*(lane>=16).
__global__ void gemm_wmma_f32(const float* __restrict__ A,
                              const float* __restrict__ B,
                              float* __restrict__ C,
                              const float* __restrict__ Csrc,
                              const float* __restrict__ bias,
                              int K, int lda, int ldb, int ldc, int ldcs,
                              long long sA, long long sB, long long sC,
                              long long sCs, int act) {
  const int lane = threadIdx.x;
  const int m0 = blockIdx.x * 16, n0 = blockIdx.y * 16;
  const long long z = blockIdx.z;
  A += z * sA; B += z * sB; C += z * sC;
  if (Csrc) Csrc += z * sCs;
  const int col = lane & 15;              // also the A-row within the tile
  const int khi = (lane >> 4) << 1;
  v8f acc = {};
  for (int kk = 0; kk < K; kk += 4) {
    v2f a, bb;
    a.x  = A[(m0 + col) * lda + kk + khi];
    a.y  = A[(m0 + col) * lda + kk + khi + 1];
    bb.x = B[(kk + khi) * ldb + n0 + col];
    bb.y = B[(kk + khi + 1) * ldb + n0 + col];
    acc = __builtin_amdgcn_wmma_f32_16x16x4_f32(false, a, false, bb,
                                                (short)0, acc, false, false);
  }
  const int half = lane >> 4;
#pragma unroll
  for (int vv = 0; vv < 8; ++vv) {
    int m = m0 + vv + (half << 3);
    int nn = n0 + col;
    float val = acc[vv];
    if (bias) val += bias[nn];
    if (act == 1) val = 0.5f * val * (1.0f + erff(val * 0.70710678118f));
    if (Csrc) val += Csrc[m * ldcs + nn];
    C[m * ldc + nn] = val;
  }
}

// -------------- P[b,i,h,f] = sum_d (q*scale + rpb) * Wrel[f, h*DK+d]  (fused
// replacement for positions@Wrel: 32x FLOP reduction)
__global__ void relP_kernel(const float* __restrict__ q,
                            const float* __restrict__ rpb,
                            const float* __restrict__ Wrel,
                            float* __restrict__ P) {
  int t = blockIdx.x * blockDim.x + threadIdx.x;
  if (t >= BATCH * NSEL * HEADS * FEAT) return;
  int f = t % FEAT;
  int h = (t / FEAT) % HEADS;
  int i = (t / (FEAT * HEADS)) % NSEL;
  int b = t / (FEAT * HEADS * NSEL);
  const float scale = 0.17677669529663687f;   // DK^-0.5
  const float* qp = q + ((size_t)b * NSEL + i) * HD + h * DK;
  const float* rp = rpb + h * DK;
  const float* wp = Wrel + f * HD + h * DK;
  float acc = 0.f;
#pragma unroll
  for (int d = 0; d < DK; ++d) acc += (qp[d] * scale + rp[d]) * wp[d];
  P[t] = acc;
}

// ---- scores: S[b,h,i,j] = (q*scale + rcb) . k  (WMMA)  +  pos(diff) . P[i,h]
__global__ void scores_kernel(const float* __restrict__ q,
                              const float* __restrict__ k,
                              const float* __restrict__ rcb,
                              const float* __restrict__ P,
                              const float* __restrict__ tab,
                              const int* __restrict__ idx,
                              float* __restrict__ S) {
  const int lane = threadIdx.x;
  const int i0 = blockIdx.x * 16, j0 = blockIdx.y * 16;
  const int z = blockIdx.z;
  const int b = z >> 2, h = z & 3;
  const float scale = 0.17677669529663687f;
  const float* qb = q + (size_t)b * NSEL * HD + h * DK;
  const float* kb = k + (size_t)b * NSEL * HD + h * DK;
  const float* cb = rcb + h * DK;
  const int col = lane & 15;
  const int khi = (lane >> 4) << 1;
  v8f acc = {};
  for (int kk = 0; kk < DK; kk += 4) {
    int d0 = kk + khi;
    v2f a, bb;
    a.x  = qb[(i0 + col) * HD + d0]     * scale + cb[d0];
    a.y  = qb[(i0 + col) * HD + d0 + 1] * scale + cb[d0 + 1];
    bb.x = kb[(j0 + col) * HD + d0];          // B[dk][j] = k[j][dk] (K^T)
    bb.y = kb[(j0 + col) * HD + d0 + 1];
    acc = __builtin_amdgcn_wmma_f32_16x16x4_f32(false, a, false, bb,
                                                (short)0, acc, false, false);
  }
  const int half = lane >> 4;
  const int* idxb = idx + b * NSEL;
  float* Sb = S + (size_t)z * NSEL * NSEL;
#pragma unroll
  for (int vv = 0; vv < 8; ++vv) {
    int i = i0 + vv + (half << 3);
    int j = j0 + col;
    int diff = idxb[i] - idxb[j] + (SEQ - 1);
    const float* pr = tab + (size_t)diff * FEAT;
    const float* Pp = P + (((size_t)b * NSEL + i) * HEADS + h) * FEAT;
    float r = 0.f;
#pragma unroll
    for (int f = 0; f < FEAT; ++f) r += pr[f] * Pp[f];
    Sb[(size_t)i * NSEL + j] = acc[vv] + r;
  }
}

// ----------------------------------------------------- row softmax over j=512
__global__ void softmax_kernel(float* __restrict__ S) {
  __shared__ float red[NSEL];
  int tid = threadIdx.x;
  size_t rowoff = (size_t)blockIdx.x * NSEL;
  float x = S[rowoff + tid];
  red[tid] = x; __syncthreads();
  for (int s = NSEL >> 1; s > 0; s >>= 1) {
    if (tid < s) red[tid] = fmaxf(red[tid], red[tid + s]);
    __syncthreads();
  }
  float mx = red[0]; __syncthreads();
  float e = expf(x - mx);
  red[tid] = e; __syncthreads();
  for (int s = NSEL >> 1; s > 0; s >>= 1) {
    if (tid < s) red[tid] += red[tid + s];
    __syncthreads();
  }
  S[rowoff + tid] = e / red[0];
}

// ------------------------------- o[b,i,h*32+d] = attn[b,h,i,:] @ v[b,:,h*32+d]
// The A operand (a 16x512 row-block of the attention matrix) is DMA'd into LDS
// with the Tensor Data Mover (2-D tile descriptor), synchronized on TENSORcnt,
// then streamed into the WMMA A layout via LDS reads (contiguous per lane).
// This replaces the worst global pattern in the program (per-lane column reads
// of S, 16 cachelines per load) with one bulk async transfer.
__global__ void attnv_kernel(const float* __restrict__ S,
                             const float* __restrict__ v,
                             float* __restrict__ o) {
  __shared__ float sA[16 * NSEL];               // 32 KB of the 320 KB/WGP LDS
  const int lane = threadIdx.x;
  const int i0 = blockIdx.x * 16, n0 = blockIdx.y * 16;
  const int z = blockIdx.z;
  const int b = z >> 2, h = z & 3;
  const float* Sb = S + (size_t)z * NSEL * NSEL + (size_t)i0 * NSEL;
  const float* vb = v + (size_t)b * NSEL * HD + h * DK;
  float* ob       = o + (size_t)b * NSEL * HD + h * DK;

  // ---- TDM descriptor (all fields uniform; built per CDNA5 ISA):
  // group0: count=1 | lds_addr (sA is the sole LDS allocation -> offset 0) |
  //         57-bit global byte address | type=2 ("image") in bits 127:126.
  // group1: data_size=2 (4 bytes), tensor 512x16, tile 512x16, dim0 stride 512.
  {
    unsigned long long ga = (unsigned long long)(uintptr_t)Sb;
    u32x4 g0;
    g0[0] = 1u;                                        // count=1, no gather
    g0[1] = 0u;                                        // lds_addr = 0
    g0[2] = (unsigned)(ga & 0xFFFFFFFFu);              // global_addr[31:0]
    g0[3] = (unsigned)((ga >> 32) & 0x01FFFFFFu)       // global_addr[56:32]
            | 0x80000000u;                             // type=2
    i32x8 g1;
    g1[0] = 0x20000;           // workgroup_mask=0, data_size=2 (4B)
    g1[1] = (int)(512u << 16); // tensor_dim0 = 512 (bits 63:48 here)
    g1[2] = (int)(16u << 16);  // tensor_dim0 hi | tensor_dim1 = 16
    g1[3] = (int)(512u << 16); // tensor_dim1 hi | tile_dim0 = 512
    g1[4] = 16;                // tile_dim1 = 16, tile_dim2 = 0
    g1[5] = 512;               // tensor_dim0_stride = 512 elements
    g1[6] = 0;
    g1[7] = 0;
    i32x4 gz4 = {0, 0, 0, 0};      // groups 2/3 unused (2-D tensor)
    i32x8 gz8 = {0, 0, 0, 0, 0, 0, 0, 0};
    __builtin_amdgcn_tensor_load_to_lds(g0, g1, gz4, gz4, gz8, 0);
    __builtin_amdgcn_s_wait_tensorcnt(0);
    asm volatile("" ::: "memory");
  }
  __syncthreads();

  const int col = lane & 15;
  const int khi = (lane >> 4) << 1;
  v8f acc = {};
  for (int kk = 0; kk < NSEL; kk += 4) {
    v2f a, bb;
    a.x  = sA[col * NSEL + kk + khi];
    a.y  = sA[col * NSEL + kk + khi + 1];
    bb.x = vb[(kk + khi) * HD + n0 + col];
    bb.y = vb[(kk + khi + 1) * HD + n0 + col];
    acc = __builtin_amdgcn_wmma_f32_16x16x4_f32(false, a, false, bb,
                                                (short)0, acc, false, false);
  }
  const int half = lane >> 4;
#pragma unroll
  for (int vv = 0; vv < 8; ++vv) {
    int m = i0 + vv + (half << 3);
    ob[(size_t)m * HD + n0 + col] = acc[vv];
  }
}

// ----------------------------------------------- scatter (last-dup wins, det.)
__global__ void scatter_out_kernel(const float* __restrict__ emb,
                                   const int* __restrict__ idx,
                                   float* __restrict__ out) {
  int t = blockIdx.x * blockDim.x + threadIdx.x;
  if (t >= BATCH * NSEL * NCH) return;
  int c = t & 31, i = (t >> 5) & 511, b = t >> 14;
  int p = idx[b * NSEL + i];
  bool last = (i == NSEL - 1) || (idx[b * NSEL + i + 1] != p);
  if (last) out[((size_t)b * NCH + c) * SEQ + p] = emb[t];
}

extern "C" void kernel_launch(void* const* d_in, const int* in_sizes, int n_in,
                              void* d_out, int out_size, void* d_ws, size_t ws_size,
                              hipStream_t stream) {
  (void)in_sizes; (void)n_in; (void)out_size; (void)ws_size;
  const float* x_skip    = (const float*)d_in[0];
  const float* attention = (const float*)d_in[1];
  const float* ln1_g = (const float*)d_in[2];
  const float* ln1_b = (const float*)d_in[3];
  const float* Wq    = (const float*)d_in[4];
  const float* Wk    = (const float*)d_in[5];
  const float* Wv    = (const float*)d_in[6];
  const float* Wrel  = (const float*)d_in[7];
  const float* rcb   = (const float*)d_in[8];
  const float* rpb   = (const float*)d_in[9];
  const float* Wo    = (const float*)d_in[10];
  const float* bo    = (const float*)d_in[11];
  const float* ln2_g = (const float*)d_in[12];
  const float* ln2_b = (const float*)d_in[13];
  const float* W1    = (const float*)d_in[14];
  const float* b1    = (const float*)d_in[15];
  const float* W2    = (const float*)d_in[16];
  const float* b2    = (const float*)d_in[17];
  float* out = (float*)d_out;

  char* w = (char*)d_ws;
  size_t off = 0;
  auto take = [&](size_t bytes) -> void* {
    void* p = w + off;
    off = (off + bytes + 255) & ~(size_t)255;
    return p;
  };
  int*   selraw = (int*)  take((size_t)BATCH * NSEL * 4);
  int*   idx    = (int*)  take((size_t)BATCH * NSEL * 4);
  float* tab    = (float*)take((size_t)TBL * FEAT * 4);
  float* emb    = (float*)take((size_t)BATCH * NSEL * NCH * 4);
  float* hbuf   = (float*)take((size_t)BATCH * NSEL * NCH * 4);
  float* qb     = (float*)take((size_t)BATCH * NSEL * HD * 4);
  float* kb     = (float*)take((size_t)BATCH * NSEL * HD * 4);
  float* vb     = (float*)take((size_t)BATCH * NSEL * HD * 4);
  float* Pb     = (float*)take((size_t)BATCH * NSEL * HEADS * FEAT * 4);
  float* Sb     = (float*)take((size_t)BATCH * HEADS * NSEL * NSEL * 4);
  float* ob     = (float*)take((size_t)BATCH * NSEL * HD * 4);
  float* h2     = (float*)take((size_t)BATCH * NSEL * NCH * 4);
  float* mb     = (float*)take((size_t)BATCH * NSEL * MLPD * 4);

  {
    int n = BATCH * NCH * SEQ;
    zero_out_kernel<<<(n + 255) / 256, 256, 0, stream>>>(out, n);
  }
  topk_rank_kernel<<<dim3((SEQ + 255) / 256, 2, BATCH), 256, 0, stream>>>(attention, selraw);
  sort_idx_kernel<<<BATCH, NSEL, 0, stream>>>(selraw, idx);
  pos_table_kernel<<<(TBL + 255) / 256, 256, 0, stream>>>(tab);
  gather_emb_kernel<<<(BATCH * NSEL * NCH + 255) / 256, 256, 0, stream>>>(x_skip, idx, emb);

  const long long sEmb = (long long)NSEL * NCH;
  const long long sQKV = (long long)NSEL * HD;
  const long long sMlp = (long long)NSEL * MLPD;

  for (int l = 0; l < 4; ++l) {
    layernorm_kernel<<<(BATCH * NSEL) / 8, 256, 0, stream>>>(
        emb, ln1_g + l * NCH, ln1_b + l * NCH, hbuf, BATCH * NSEL);

    // q/k/v: [512,32] @ [32,128] per batch
    gemm_wmma_f32<<<dim3(NSEL / 16, HD / 16, BATCH), 32, 0, stream>>>(
        hbuf, Wq + (size_t)l * NCH * HD, qb, nullptr, nullptr,
        NCH, NCH, HD, HD, 0, sEmb, 0, sQKV, 0, 0);
    gemm_wmma_f32<<<dim3(NSEL / 16, HD / 16, BATCH), 32, 0, stream>>>(
        hbuf, Wk + (size_t)l * NCH * HD, kb, nullptr, nullptr,
        NCH, NCH, HD, HD, 0, sEmb, 0, sQKV, 0, 0);
    gemm_wmma_f32<<<dim3(NSEL / 16, HD / 16, BATCH), 32, 0, stream>>>(
        hbuf, Wv + (size_t)l * NCH * HD, vb, nullptr, nullptr,
        NCH, NCH, HD, HD, 0, sEmb, 0, sQKV, 0, 0);

    relP_kernel<<<(BATCH * NSEL * HEADS * FEAT + 255) / 256, 256, 0, stream>>>(
        qb, rpb + l * HEADS * DK, Wrel + (size_t)l * FEAT * HD, Pb);

    scores_kernel<<<dim3(NSEL / 16, NSEL / 16, BATCH * HEADS), 32, 0, stream>>>(
        qb, kb, rcb + l * HEADS * DK, Pb, tab, idx, Sb);

    softmax_kernel<<<BATCH * HEADS * NSEL, NSEL, 0, stream>>>(Sb);

    attnv_kernel<<<dim3(NSEL / 16, DK / 16, BATCH * HEADS), 32, 0, stream>>>(Sb, vb, ob);

    // emb += o @ Wo + bo   ([512,128] @ [128,32])
    gemm_wmma_f32<<<dim3(NSEL / 16, NCH / 16, BATCH), 32, 0, stream>>>(
        ob, Wo + (size_t)l * HD * NCH, emb, emb, bo + l * NCH,
        HD, HD, NCH, NCH, NCH, sQKV, 0, sEmb, sEmb, 0);

    layernorm_kernel<<<(BATCH * NSEL) / 8, 256, 0, stream>>>(
        emb, ln2_g + l * NCH, ln2_b + l * NCH, h2, BATCH * NSEL);

    // m = gelu(h2 @ W1 + b1)   ([512,32] @ [32,512])
    gemm_wmma_f32<<<dim3(NSEL / 16, MLPD / 16, BATCH), 32, 0, stream>>>(
        h2, W1 + (size_t)l * NCH * MLPD, mb, nullptr, b1 + l * MLPD,
        NCH, NCH, MLPD, MLPD, 0, sEmb, 0, sMlp, 0, 1);

    // emb += m @ W2 + b2   ([512,512] @ [512,32])
    gemm_wmma_f32<<<dim3(NSEL / 16, NCH / 16, BATCH), 32, 0, stream>>>(
        mb, W2 + (size_t)l * MLPD * NCH, emb, emb, b2 + l * NCH,
        MLPD, MLPD, NCH, NCH, NCH, sMlp, 0, sEmb, sEmb, 0);
  }

  scatter_out_kernel<<<(BATCH * NSEL * NCH + 255) / 256, 256, 0, stream>>>(emb, idx, out);
}